// ICTDIrrepsE3Conv_26628797235369
// MI455X (gfx1250) — compile-verified
//
#include <hip/hip_runtime.h>
#include <math.h>

// ---------------------------------------------------------------------------
// Types / helpers
// ---------------------------------------------------------------------------
typedef float v2f __attribute__((ext_vector_type(2)));
typedef float v8f __attribute__((ext_vector_type(8)));

#define NPATH 15
#define NCG   615   // total dense CG elements over all paths

// Path tables (compile-time; order matches the Python PATHS enumeration)
constexpr int cPL1[NPATH]  = {0,0,0,1,1,1,1,1,1,2,2,2,2,2,2};
constexpr int cPL2[NPATH]  = {0,1,2,0,1,1,1,2,2,0,1,1,2,2,2};
constexpr int cPL3[NPATH]  = {0,1,2,1,0,1,2,1,2,2,1,2,0,1,2};
constexpr int cCGOFF[NPATH+1] = {0,1,10,35,44,53,80,125,170,245,270,315,390,415,490,615};
constexpr int cYB[3] = {0,1,4};          // Y base offset per l
constexpr int cGB[3] = {0,16,64};        // output base per l3 block
// ALPHA[l3] = 1/sqrt(8 * n_paths_with_that_l3):  l3=0 -> 3 paths, l3=1/2 -> 6
constexpr float cALPHA[3] = {0.2041241452319315f, 0.1443375672974065f, 0.1443375672974065f};

// silu via hardware reciprocal (v_rcp_f32, ~1ulp) instead of IEEE division:
// avoids the v_div_scale/v_div_fmas/v_div_fixup sequence in the hot path.
__device__ __forceinline__ float silu(float x) {
  float t = 1.0f + __expf(-x);
  return x * __builtin_amdgcn_rcpf(t);
}

__device__ __forceinline__ v8f wmma4(v2f a, v2f b, v8f c) {
  // V_WMMA_F32_16X16X4_F32 : D = A(16x4) * B(4x16) + C(16x16), fp32
  return __builtin_amdgcn_wmma_f32_16x16x4_f32(false, a, false, b, (short)0, c,
                                               false, false);
}

// ---------------------------------------------------------------------------
// K0: Clebsch-Gordan init (exact port of _cg/_q/_real_cg in double precision)
// ---------------------------------------------------------------------------
__device__ const int gEOFF[NPATH+1] = {0,1,10,35,44,53,80,125,170,245,270,315,390,415,490,615};
__device__ const int gL1[NPATH] = {0,0,0,1,1,1,1,1,1,2,2,2,2,2,2};
__device__ const int gL2[NPATH] = {0,1,2,0,1,1,1,2,2,0,1,1,2,2,2};
__device__ const int gL3[NPATH] = {0,1,2,1,0,1,2,1,2,2,1,2,0,1,2};

struct cd { double re, im; };
__device__ __forceinline__ cd cmul(cd a, cd b) {
  cd r; r.re = a.re*b.re - a.im*b.im; r.im = a.re*b.im + a.im*b.re; return r;
}
__device__ double dfact(int n) { double r = 1.0; for (int i = 2; i <= n; ++i) r *= (double)i; return r; }

__device__ double cgcoef(int j1,int j2,int j3,int m1,int m2,int m3) {
  if (m1 + m2 != m3) return 0.0;
  double pre = sqrt((2.0*j3+1.0)*dfact(j3+j1-j2)*dfact(j3-j1+j2)*dfact(j1+j2-j3)/dfact(j1+j2+j3+1));
  pre *= sqrt(dfact(j3+m3)*dfact(j3-m3)*dfact(j1-m1)*dfact(j1+m1)*dfact(j2-m2)*dfact(j2+m2));
  double s = 0.0;
  for (int k = 0; k <= j1+j2-j3; ++k) {
    int d0=k, d1=j1+j2-j3-k, d2=j1-m1-k, d3=j2+m2-k, d4=j3-j2+m1+k, d5=j3-j1-m2+k;
    if (d0<0||d1<0||d2<0||d3<0||d4<0||d5<0) continue;
    double term = 1.0/(dfact(d0)*dfact(d1)*dfact(d2)*dfact(d3)*dfact(d4)*dfact(d5));
    s += (k & 1) ? -term : term;
  }
  return pre * s;
}

// q-matrix element q[l](row a, col), times (-i)^l  (port of _q)
__device__ cd qval(int l, int a, int col) {
  const double INV = 0.70710678118654752440;
  int m = a - l;
  cd v; v.re = 0.0; v.im = 0.0;
  if (m < 0) {
    if (col == 2*l - a)      { v.re = INV; }
    else if (col == a)       { v.im = -INV; }
  } else if (m == 0) {
    if (col == l)            { v.re = 1.0; }
  } else {
    double s = (m & 1) ? -1.0 : 1.0;
    if (col == a)            { v.re = s * INV; }
    else if (col == 2*l - a) { v.im = s * INV; }
  }
  if (l == 1) { double t = v.re; v.re = v.im; v.im = -t; }   // * (-i)
  else if (l == 2) { v.re = -v.re; v.im = -v.im; }           // * (-i)^2 = -1
  return v;
}

__global__ void cg_init_kernel(float* __restrict__ cgDense) {
  __shared__ double sre[NCG], sim[NCG];
  __shared__ int sPick[NPATH];
  int tid = threadIdx.x;
  for (int t = tid; t < NCG; t += blockDim.x) {
    int p = 0; while (t >= gEOFF[p+1]) ++p;
    int l1 = gL1[p], l2 = gL2[p], l3 = gL3[p];
    int n1 = 2*l1+1, n2 = 2*l2+1, nk = 2*l3+1;
    int rel = t - gEOFF[p];
    int k = rel % nk, j = (rel/nk) % n2, i = rel/(nk*n2);
    cd sum; sum.re = 0.0; sum.im = 0.0;
    for (int a = 0; a < n1; ++a)
      for (int b = 0; b < n2; ++b)
        for (int c = 0; c < nk; ++c) {
          double C = cgcoef(l1, l2, l3, a-l1, b-l2, c-l3);
          if (C == 0.0) continue;
          cd t1 = qval(l1, a, i), t2 = qval(l2, b, j), t3 = qval(l3, c, k);
          t3.im = -t3.im;                       // conj(q(l3))
          cd pr = cmul(cmul(t1, t2), t3);
          sum.re += pr.re * C; sum.im += pr.im * C;
        }
    sre[t] = sum.re; sim[t] = sum.im;
  }
  __syncthreads();
  if (tid == 0) {
    for (int p = 0; p < NPATH; ++p) {
      double ar = 0.0, ai = 0.0;
      for (int t = gEOFF[p]; t < gEOFF[p+1]; ++t) { ar += fabs(sre[t]); ai += fabs(sim[t]); }
      sPick[p] = (ar >= ai) ? 1 : 0;
    }
  }
  __syncthreads();
  for (int t = tid; t < NCG; t += blockDim.x) {
    int p = 0; while (t >= gEOFF[p+1]) ++p;
    cgDense[t] = (float)(sPick[p] ? sre[t] : sim[t]);
  }
}

// ---------------------------------------------------------------------------
// K1: per-node embedding MLP  Ai = silu(emb_table[A] @ W1 + b1) @ W2 + b2
// ---------------------------------------------------------------------------
__global__ __launch_bounds__(128) void node_mlp_kernel(
    const int* __restrict__ A, const float* __restrict__ emb_table,
    const float* __restrict__ w1, const float* __restrict__ b1,
    const float* __restrict__ w2, const float* __restrict__ b2,
    float* __restrict__ Ai, int N) {
  int n = blockIdx.x * blockDim.x + threadIdx.x;
  if (n >= N) return;
  int a = A[n];
  float e[16];
  const float* ev = emb_table + (size_t)a * 16;
#pragma unroll
  for (int i = 0; i < 16; ++i) e[i] = ev[i];
  float h[64];
#pragma unroll 8
  for (int j = 0; j < 64; ++j) {
    float s = b1[j];
#pragma unroll
    for (int i = 0; i < 16; ++i) s = fmaf(e[i], w1[i*64 + j], s);
    h[j] = silu(s);
  }
#pragma unroll
  for (int o = 0; o < 8; ++o) {
    float s = b2[o];
#pragma unroll 8
    for (int j = 0; j < 64; ++j) s = fmaf(h[j], w2[j*8 + o], s);
    Ai[(size_t)n*8 + o] = s;
  }
}

// ---------------------------------------------------------------------------
// K2: fused edge kernel (WMMA radial MLP + tensor product + atomic scatter)
// 128 threads = 4 waves, 32 edges per wave, 128 edges per block.
// ---------------------------------------------------------------------------
#define HSTRIDE 65   // 32x64 staging with padding (bank-conflict friendly)
#define GSTRIDE 17   // 32x16 gates staging

__global__ __launch_bounds__(128) void edge_kernel(
    const float* __restrict__ pos, const int* __restrict__ batch,
    const int* __restrict__ esrc, const int* __restrict__ edst,
    const float* __restrict__ shifts, const float* __restrict__ cell,
    const float* __restrict__ fw1, const float* __restrict__ fb1,
    const float* __restrict__ fw2, const float* __restrict__ fb2,
    const float* __restrict__ fw3, const float* __restrict__ fb3,
    const float* __restrict__ tpw, const float* __restrict__ Ai,
    const float* __restrict__ cgDense, float* __restrict__ out,
    int E, float invAvg) {
  __shared__ float sCG[NCG];
  __shared__ float sTP[NPATH * 8 * 16];
  __shared__ float sW3[64 * 16];     // fc_w3 zero-padded 15 -> 16 cols
  __shared__ float sB3[16];
  __shared__ float sStage[4][32 * HSTRIDE];

  const int tid = threadIdx.x;
  // cooperative loads of broadcast tables
  for (int i = tid; i < NCG; i += 128) sCG[i] = cgDense[i];
  for (int i = tid; i < NPATH*8*16; i += 128) sTP[i] = tpw[i];
  for (int i = tid; i < 64*16; i += 128) {
    int k = i >> 4, n = i & 15;
    sW3[i] = (n < 15) ? fw3[k*15 + n] : 0.0f;
  }
  if (tid < 16) sB3[tid] = (tid < 15) ? fb3[tid] : 0.0f;
  __syncthreads();

  const int wave = tid >> 5, lane = tid & 31;
  const int hi = lane >> 4, lo = lane & 15;
  float* hS = &sStage[wave][0];

  const int e = blockIdx.x * 128 + tid;
  const bool valid = (e < E);
  const int ec = valid ? e : (E - 1);

  // ---- geometry ----
  const int src = esrc[ec], dst = edst[ec];
  const int ci = batch[src];
  const float sh0 = shifts[(size_t)ec*3+0], sh1 = shifts[(size_t)ec*3+1], sh2 = shifts[(size_t)ec*3+2];
  const float* C = cell + (size_t)ci * 9;
  float vx = pos[(size_t)dst*3+0] - pos[(size_t)src*3+0] + sh0*C[0] + sh1*C[3] + sh2*C[6];
  float vy = pos[(size_t)dst*3+1] - pos[(size_t)src*3+1] + sh0*C[1] + sh1*C[4] + sh2*C[7];
  float vz = pos[(size_t)dst*3+2] - pos[(size_t)src*3+2] + sh0*C[2] + sh1*C[5] + sh2*C[8];
  const float len = sqrtf(vx*vx + vy*vy + vz*vz);
  const float rinv = 1.0f / fmaxf(len, 1e-8f);
  const float nx = vx*rinv, ny = vy*rinv, nz = vz*rinv;

  const float S3 = 1.7320508075688772f, S15 = 3.872983346207417f, S5 = 2.23606797749979f;
  float Y[9];
  Y[0] = 1.0f;
  Y[1] = S3*ny; Y[2] = S3*nz; Y[3] = S3*nx;
  Y[4] = S15*nx*ny; Y[5] = S15*ny*nz; Y[6] = 0.5f*S5*(3.0f*nz*nz - 1.0f);
  Y[7] = S15*nx*nz; Y[8] = 0.5f*S15*(nx*nx - ny*ny);

  // ---- radial basis: soft one-hot * sqrt(8) ----
  const float step = 2.0f / 9.0f;
  const float emb_scale = 2.8284271247461903f / 1.12f;
#pragma unroll
  for (int i = 0; i < 8; ++i) {
    float d = (len - (float)(i+1)*step) / step;
    hS[lane*HSTRIDE + i] = __expf(-d*d) * emb_scale;   // stage emb row
  }
  __builtin_amdgcn_wave_barrier();

  const v8f vzero = {0.f,0.f,0.f,0.f,0.f,0.f,0.f,0.f};

  // ---- Layer 1: [32x8] @ fw1[8x64] -> silu -> hS[32x64]  (WMMA f32 16x16x4)
  v2f a1[2][2];  // [mtile][kblock]
#pragma unroll
  for (int mt = 0; mt < 2; ++mt)
#pragma unroll
    for (int kb = 0; kb < 2; ++kb) {
      int m = lo + 16*mt, kk = kb*4 + 2*hi;
      a1[mt][kb].x = hS[m*HSTRIDE + kk];
      a1[mt][kb].y = hS[m*HSTRIDE + kk + 1];
    }
#pragma unroll
  for (int nt = 0; nt < 4; ++nt) {
    v8f acc0 = vzero, acc1 = vzero;
#pragma unroll
    for (int kb = 0; kb < 2; ++kb) {
      int kk = kb*4 + 2*hi, col = nt*16 + lo;
      v2f b; b.x = fw1[kk*64 + col]; b.y = fw1[(kk+1)*64 + col];
      acc0 = wmma4(a1[0][kb], b, acc0);
      acc1 = wmma4(a1[1][kb], b, acc1);
    }
    float bias = fb1[nt*16 + lo];
#pragma unroll
    for (int r = 0; r < 8; ++r) {
      hS[(r + 8*hi)      * HSTRIDE + nt*16 + lo] = silu(acc0[r] + bias);
      hS[(r + 8*hi + 16) * HSTRIDE + nt*16 + lo] = silu(acc1[r] + bias);
    }
  }
  __builtin_amdgcn_wave_barrier();

  // ---- Layer 2: [32x64] @ fw2[64x64] -> silu -> hS ----
  v8f d2[4][2];
#pragma unroll
  for (int nt = 0; nt < 4; ++nt) { d2[nt][0] = vzero; d2[nt][1] = vzero; }
#pragma unroll
  for (int kk = 0; kk < 16; ++kk) {
    int kb4 = kk*4 + 2*hi;
    v2f a0, am;
    a0.x = hS[lo*HSTRIDE + kb4];        a0.y = hS[lo*HSTRIDE + kb4 + 1];
    am.x = hS[(lo+16)*HSTRIDE + kb4];   am.y = hS[(lo+16)*HSTRIDE + kb4 + 1];
#pragma unroll
    for (int nt = 0; nt < 4; ++nt) {
      int col = nt*16 + lo;
      v2f b; b.x = fw2[kb4*64 + col]; b.y = fw2[(kb4+1)*64 + col];
      d2[nt][0] = wmma4(a0, b, d2[nt][0]);
      d2[nt][1] = wmma4(am, b, d2[nt][1]);
    }
  }
  __builtin_amdgcn_wave_barrier();
#pragma unroll
  for (int nt = 0; nt < 4; ++nt) {
    float bias = fb2[nt*16 + lo];
#pragma unroll
    for (int r = 0; r < 8; ++r) {
      hS[(r + 8*hi)      * HSTRIDE + nt*16 + lo] = silu(d2[nt][0][r] + bias);
      hS[(r + 8*hi + 16) * HSTRIDE + nt*16 + lo] = silu(d2[nt][1][r] + bias);
    }
  }
  __builtin_amdgcn_wave_barrier();

  // ---- Layer 3: gates = h2 @ fw3[64x15] + b3  (padded to 16 cols) ----
  v8f d3a = vzero, d3b = vzero;
#pragma unroll
  for (int kk = 0; kk < 16; ++kk) {
    int kb4 = kk*4 + 2*hi;
    v2f a0, am;
    a0.x = hS[lo*HSTRIDE + kb4];        a0.y = hS[lo*HSTRIDE + kb4 + 1];
    am.x = hS[(lo+16)*HSTRIDE + kb4];   am.y = hS[(lo+16)*HSTRIDE + kb4 + 1];
    v2f b; b.x = sW3[kb4*16 + lo]; b.y = sW3[(kb4+1)*16 + lo];
    d3a = wmma4(a0, b, d3a);
    d3b = wmma4(am, b, d3b);
  }
  __builtin_amdgcn_wave_barrier();
  {
    float bb = sB3[lo];
#pragma unroll
    for (int r = 0; r < 8; ++r) {
      hS[(r + 8*hi)      * GSTRIDE + lo] = d3a[r] + bb;
      hS[(r + 8*hi + 16) * GSTRIDE + lo] = d3b[r] + bb;
    }
  }
  __builtin_amdgcn_wave_barrier();
  float gates[NPATH];
#pragma unroll
  for (int p = 0; p < NPATH; ++p) gates[p] = hS[lane*GSTRIDE + p];

  // ---- per-node features for this edge's source ----
  float ai[8];
  const float* ap = Ai + (size_t)src * 8;
#pragma unroll
  for (int i = 0; i < 8; ++i) ai[i] = ap[i];

  // ---- tensor product, staged by l3 to cap register pressure ----
#pragma unroll
  for (int g = 0; g < 3; ++g) {
    const int nk = 2*g + 1;
    float acc[80];
#pragma unroll
    for (int t = 0; t < 16*5; ++t) if (t < 16*nk) acc[t] = 0.0f;
#pragma unroll
    for (int p = 0; p < NPATH; ++p) {
      if (cPL3[p] != g) continue;
      const int l1 = cPL1[p], l2 = cPL2[p];
      const int n1 = 2*l1 + 1, n2 = 2*l2 + 1;
      float geo[5];
#pragma unroll
      for (int k = 0; k < 5; ++k) if (k < nk) geo[k] = 0.0f;
#pragma unroll
      for (int m = 0; m < 5; ++m) {
        if (m >= n1) continue;
#pragma unroll
        for (int n = 0; n < 5; ++n) {
          if (n >= n2) continue;
          float yy = Y[cYB[l1] + m] * Y[cYB[l2] + n];
#pragma unroll
          for (int k = 0; k < 5; ++k)
            if (k < nk) geo[k] = fmaf(yy, sCG[cCGOFF[p] + (m*n2 + n)*nk + k], geo[k]);
        }
      }
      const float s = gates[p] * cALPHA[g] * invAvg;
#pragma unroll
      for (int k = 0; k < 5; ++k) if (k < nk) geo[k] *= s;
#pragma unroll
      for (int c = 0; c < 16; ++c) {
        float ch = 0.0f;
#pragma unroll
        for (int i = 0; i < 8; ++i) ch = fmaf(ai[i], sTP[(p*8 + i)*16 + c], ch);
#pragma unroll
        for (int k = 0; k < 5; ++k)
          if (k < nk) acc[c*nk + k] = fmaf(ch, geo[k], acc[c*nk + k]);
      }
    }
    if (valid) {
      float* op = out + (size_t)dst * 144 + cGB[g];
#pragma unroll
      for (int t = 0; t < 16*5; ++t)
        if (t < 16*nk)
          __hip_atomic_fetch_add(&op[t], acc[t], __ATOMIC_RELAXED, __HIP_MEMORY_SCOPE_AGENT);
    }
  }
}

// ---------------------------------------------------------------------------
// Host launcher
// ---------------------------------------------------------------------------
extern "C" void kernel_launch(void* const* d_in, const int* in_sizes, int n_in,
                              void* d_out, int out_size, void* d_ws, size_t ws_size,
                              hipStream_t stream) {
  const float* pos       = (const float*)d_in[0];
  const int*   A         = (const int*)  d_in[1];
  const int*   batch     = (const int*)  d_in[2];
  const int*   esrc      = (const int*)  d_in[3];
  const int*   edst      = (const int*)  d_in[4];
  const float* shifts    = (const float*)d_in[5];
  const float* cell      = (const float*)d_in[6];
  const float* emb_table = (const float*)d_in[7];
  const float* mw1 = (const float*)d_in[8];
  const float* mb1 = (const float*)d_in[9];
  const float* mw2 = (const float*)d_in[10];
  const float* mb2 = (const float*)d_in[11];
  const float* fw1 = (const float*)d_in[12];
  const float* fb1 = (const float*)d_in[13];
  const float* fw2 = (const float*)d_in[14];
  const float* fb2 = (const float*)d_in[15];
  const float* fw3 = (const float*)d_in[16];
  const float* fb3 = (const float*)d_in[17];
  const float* tpw = (const float*)d_in[18];

  const int N = in_sizes[0] / 3;
  const int E = in_sizes[3];

  float avg = (float)E / (float)N;
  if (avg < 1e-8f) avg = 1e-8f;
  const float invAvg = 1.0f / avg;

  // workspace layout: [0,4096): dense CG table; [4096, ...): Ai[N*8]
  float* cgD = (float*)d_ws;
  float* Ai  = (float*)((char*)d_ws + 4096);

  hipMemsetAsync(d_out, 0, (size_t)out_size * sizeof(float), stream);

  cg_init_kernel<<<1, 128, 0, stream>>>(cgD);
  node_mlp_kernel<<<(N + 127) / 128, 128, 0, stream>>>(A, emb_table, mw1, mb1, mw2, mb2, Ai, N);
  edge_kernel<<<(E + 127) / 128, 128, 0, stream>>>(
      pos, batch, esrc, edst, shifts, cell,
      fw1, fb1, fw2, fb2, fw3, fb3, tpw, Ai, cgD,
      (float*)d_out, E, invAvg);
}